// AttMoE_24678882083377
// MI455X (gfx1250) — compile-verified
//
#include <hip/hip_runtime.h>
#include <math.h>

// Problem constants (from reference)
#define B_  16
#define S_  1024
#define F_  64
#define H_  512
#define NH_ 4
#define E_  12
#define DH_ 128

typedef __attribute__((ext_vector_type(2))) float v2f;
typedef __attribute__((ext_vector_type(8))) float v8f;

// CDNA5 fp32 matrix core: D(16x16,f32) = A(16x4,f32) x B(4x16,f32) + C
static __device__ __forceinline__ v8f wmma4(v2f a, v2f b, v8f c) {
  return __builtin_amdgcn_wmma_f32_16x16x4_f32(false, a, false, b, (short)0, c,
                                               false, false);
}

// gfx1250 async copy: global -> LDS, 16B per lane, tracked by ASYNCcnt.
static __device__ __forceinline__ void async_g2l_b128(const void* lds_dst,
                                                      const void* gsrc) {
  unsigned ldsoff = (unsigned)(unsigned long long)lds_dst;  // low 32b = LDS addr
  unsigned long long g = (unsigned long long)gsrc;
  asm volatile("global_load_async_to_lds_b128 %0, %1, off"
               :
               : "v"(ldsoff), "v"(g)
               : "memory");
}
static __device__ __forceinline__ void wait_async0() {
  asm volatile("s_wait_asynccnt 0x0" ::: "memory");
}

// ---------------------------------------------------------------------------
// WMMA GEMM (compile-time K, N): C[M,N] = A[M,K] @ B[K,N] + bias[N]
// 8 waves/block share one 16-row A tile (async-staged to LDS, padded pitch
// 516%64==4 -> conflict-free ds_load_b64 fragments). Each wave owns one 16x16
// output tile; constant strides let B-loads/C-stores use immediate offsets.
// ---------------------------------------------------------------------------
template <int K, int N>
__global__ __launch_bounds__(256) void gemm16(
    const float* __restrict__ A, const float* __restrict__ Bm,
    const float* __restrict__ bias, float* __restrict__ C, int M) {
  __shared__ float ldsA[16][516];
  const int lane = threadIdx.x & 31;
  const int wv   = threadIdx.x >> 5;
  constexpr int tilesN = N >> 4;
  const int wid0 = blockIdx.x * 8;              // block-aligned tile group
  const int tm = (wid0 / tilesN) << 4;          // same row tile for all 8 waves
  const int tn = ((wid0 % tilesN) + wv) << 4;
  const int half = lane >> 4, l = lane & 15;

  // Stage A[tm:tm+16, 0:K] -> LDS, coalesced 16B chunks (K/4 chunks per row).
  constexpr int chunks = (16 * K) >> 2;
#pragma unroll
  for (int i = 0; i < chunks / 256; ++i) {
    const int c = threadIdx.x + i * 256;
    const int row = c / (K >> 2);               // power-of-two divide
    const int col = c % (K >> 2);
    async_g2l_b128(&ldsA[row][col * 4], A + (size_t)(tm + row) * K + col * 4);
  }
  wait_async0();
  __syncthreads();

  v8f acc = {0.f, 0.f, 0.f, 0.f, 0.f, 0.f, 0.f, 0.f};
  const float* pb = Bm + (size_t)(half * 2) * N + tn + l;
#pragma unroll 8
  for (int k0 = 0; k0 < K; k0 += 4) {
    v2f a; a.x = ldsA[l][k0 + half * 2]; a.y = ldsA[l][k0 + half * 2 + 1];
    v2f b; b.x = pb[(size_t)k0 * N]; b.y = pb[(size_t)(k0 + 1) * N];
    acc = wmma4(a, b, acc);
  }
  const int col = tn + l;
  const float bval = bias[col];
  float* pc = C + (size_t)(tm + half * 8) * N + col;
#pragma unroll
  for (int r = 0; r < 8; ++r) pc[(size_t)r * N] = acc[r] + bval;
}

// ---------------------------------------------------------------------------
// Fused flash attention. Block = 4 waves = 4 query tiles of the SAME (b,head);
// K/V 16x128 tiles are async-staged to LDS (double-buffered) and shared by all
// 4 waves. Q lives in registers as pre-formatted A fragments for all 64 key
// blocks. P is relaid C-layout -> A-layout through a wave-private LDS tile.
// ---------------------------------------------------------------------------
__global__ __launch_bounds__(128) void attn_flash(
    const float* __restrict__ Q, const float* __restrict__ Kt,
    const float* __restrict__ V, float* __restrict__ O) {
  __shared__ float ldsK[2][16][132];  // 132%64==4 -> conflict-free frag reads
  __shared__ float ldsV[2][16][132];
  __shared__ float ldsP[4][16 * 17];
  const int lane  = threadIdx.x & 31;
  const int wslot = threadIdx.x >> 5;
  const int bh = blockIdx.x >> 4;               // 16 blocks per (b,h)
  const int qt = ((blockIdx.x & 15) << 2) + wslot;
  const int h  = bh & (NH_ - 1);
  const int b  = bh >> 2;
  const float* Qb = Q + (size_t)b * S_ * H_ + h * DH_;
  const float* Kb = Kt + (size_t)b * S_ * H_ + h * DH_;
  const float* Vb = V + (size_t)b * S_ * H_ + h * DH_;
  float*       Ob = O + (size_t)b * S_ * H_ + h * DH_;
  const int qr0  = qt << 4;
  const int half = lane >> 4, l = lane & 15;
  float* myP = ldsP[wslot];
  const float scale = 0.08838834764831845f;  // 1/sqrt(128)

  // ---- Q tile -> registers as 32 A-fragments (reused for all key blocks) ----
  v2f qa[DH_ / 4];
#pragma unroll
  for (int kk = 0; kk < DH_ / 4; ++kk) {
    const float* pa = Qb + (size_t)(qr0 + l) * H_ + kk * 4 + half * 2;
    qa[kk].x = pa[0];
    qa[kk].y = pa[1];
  }

  // Cooperative stage of one 16x128 K tile + V tile (8KB each, 16B chunks).
  auto stage = [&](int j0, int buf) {
#pragma unroll
    for (int i = 0; i < 4; ++i) {
      const int c = threadIdx.x + i * 128;  // 512 chunks per matrix
      const int row = c >> 5;               // 32 chunks per 128-float row
      const int col = c & 31;
      async_g2l_b128(&ldsK[buf][row][col * 4],
                     Kb + (size_t)(j0 + row) * H_ + col * 4);
      async_g2l_b128(&ldsV[buf][row][col * 4],
                     Vb + (size_t)(j0 + row) * H_ + col * 4);
    }
  };

  v8f acc[8];
#pragma unroll
  for (int n = 0; n < 8; ++n) acc[n] = (v8f){0.f,0.f,0.f,0.f,0.f,0.f,0.f,0.f};
  float mrun[8], lrun[8];
#pragma unroll
  for (int r = 0; r < 8; ++r) { mrun[r] = -INFINITY; lrun[r] = 0.f; }

  stage(0, 0);
  for (int jb = 0; jb < S_ / 16; ++jb) {
    wait_async0();       // tile jb landed (own wave's asyncs)
    __syncthreads();     // ...and everyone else's; also: all done reading jb-1
    if (jb + 1 < S_ / 16) stage((jb + 1) * 16, (jb + 1) & 1);  // prefetch
    const int buf = jb & 1;

    // ---- S = Q @ K_tile^T ----
    v8f s = {0.f,0.f,0.f,0.f,0.f,0.f,0.f,0.f};
#pragma unroll
    for (int kk = 0; kk < DH_ / 4; ++kk) {
      v2f bf;
      bf.x = ldsK[buf][l][kk * 4 + half * 2];
      bf.y = ldsK[buf][l][kk * 4 + half * 2 + 1];
      s = wmma4(qa[kk], bf, s);
    }
    // ---- online softmax (row stats replicated across each 16-lane half) ----
#pragma unroll
    for (int r = 0; r < 8; ++r) {
      float sv = s[r] * scale;
      float rm = sv;
      rm = fmaxf(rm, __shfl_xor(rm, 1));
      rm = fmaxf(rm, __shfl_xor(rm, 2));
      rm = fmaxf(rm, __shfl_xor(rm, 4));
      rm = fmaxf(rm, __shfl_xor(rm, 8));
      const float mnew = fmaxf(mrun[r], rm);
      const float sc   = __expf(mrun[r] - mnew);
      const float p    = __expf(sv - mnew);
      float ps = p;
      ps += __shfl_xor(ps, 1);
      ps += __shfl_xor(ps, 2);
      ps += __shfl_xor(ps, 4);
      ps += __shfl_xor(ps, 8);
      lrun[r] = lrun[r] * sc + ps;
      mrun[r] = mnew;
#pragma unroll
      for (int n = 0; n < 8; ++n) acc[n][r] *= sc;
      myP[(r + half * 8) * 17 + l] = p;  // C-layout -> LDS row-major 16x16
    }
    // ---- acc += P @ V_tile ----
#pragma unroll
    for (int kk = 0; kk < 16; kk += 4) {
      v2f a;
      a.x = myP[l * 17 + kk + half * 2];
      a.y = myP[l * 17 + kk + half * 2 + 1];
#pragma unroll
      for (int n = 0; n < 8; ++n) {
        v2f bf;
        bf.x = ldsV[buf][kk + half * 2][n * 16 + l];
        bf.y = ldsV[buf][kk + half * 2 + 1][n * 16 + l];
        acc[n] = wmma4(a, bf, acc[n]);
      }
    }
  }
  // ---- normalize, store ----
#pragma unroll
  for (int r = 0; r < 8; ++r) {
    const float inv = 1.f / lrun[r];
    const int row = qr0 + r + half * 8;
#pragma unroll
    for (int n = 0; n < 8; ++n)
      Ob[(size_t)row * H_ + n * 16 + l] = acc[n][r] * inv;
  }
}

// ---------------------------------------------------------------------------
// Fold pre-projection into MHA in-projection: Wc = W1@W2, bc = b1@W2 + b2
// ---------------------------------------------------------------------------
__global__ void combine_wb(const float* __restrict__ W1, const float* __restrict__ W2,
                           const float* __restrict__ b1, const float* __restrict__ b2,
                           float* __restrict__ Wc, float* __restrict__ bc) {
  const int t = blockIdx.x * blockDim.x + threadIdx.x;
  if (t < F_ * H_) {
    const int f = t / H_, hc = t % H_;
    float s = 0.f;
    for (int d = 0; d < H_; ++d) s += W1[f * H_ + d] * W2[d * H_ + hc];
    Wc[t] = s;
  } else if (t < F_ * H_ + H_) {
    const int hc = t - F_ * H_;
    float s = b2[hc];
    for (int d = 0; d < H_; ++d) s += b1[d] * W2[d * H_ + hc];
    bc[hc] = s;
  }
}

// Fold experts through the head: weh[e,d] = sum_h We[e,d,h]*Wh[h]; beh[e]=be[e]·Wh
__global__ void fold_expert(const float* __restrict__ We, const float* __restrict__ be,
                            const float* __restrict__ Wh, float* __restrict__ weh,
                            float* __restrict__ beh) {
  const int t = blockIdx.x * blockDim.x + threadIdx.x;
  if (t < E_ * H_) {
    const int e = t / H_, d = t % H_;
    const float* base = We + ((size_t)e * H_ + d) * H_;
    float s = 0.f;
    for (int hh = 0; hh < H_; ++hh) s += base[hh] * Wh[hh];
    weh[t] = s;
  } else if (t < E_ * H_ + E_) {
    const int e = t - E_ * H_;
    float s = 0.f;
    for (int hh = 0; hh < H_; ++hh) s += be[e * H_ + hh] * Wh[hh];
    beh[e] = s;
  }
}

__global__ void init_out(float* __restrict__ out, const float* __restrict__ bh) {
  if (threadIdx.x < B_) out[threadIdx.x] = bh[0];
}

// ---------------------------------------------------------------------------
// Gating + pooled reduction: one wave per token.
// out[b] += (1/S) * sum_e g_e * (o2 . weh[e] + beh[e])
// ---------------------------------------------------------------------------
__global__ __launch_bounds__(256) void moe_gate_reduce(
    const float* __restrict__ o2, const float* __restrict__ Wg,
    const float* __restrict__ weh, const float* __restrict__ beh,
    float* __restrict__ out) {
  const int lane = threadIdx.x & 31;
  const int tok  = blockIdx.x * (blockDim.x >> 5) + (threadIdx.x >> 5);
  const float* ob = o2 + (size_t)tok * H_ + lane * 16;
  float c[16];
#pragma unroll
  for (int i = 0; i < 16; ++i) c[i] = ob[i];

  float z[E_];
#pragma unroll
  for (int e = 0; e < E_; ++e) {
    float p = 0.f;
#pragma unroll
    for (int i = 0; i < 16; ++i) p += c[i] * Wg[(lane * 16 + i) * E_ + e];
    p += __shfl_xor(p, 16);
    p += __shfl_xor(p, 8);
    p += __shfl_xor(p, 4);
    p += __shfl_xor(p, 2);
    p += __shfl_xor(p, 1);
    z[e] = p;
  }
  // top-2 (ties -> lowest index, matching lax.top_k)
  int e1 = 0;
#pragma unroll
  for (int e = 1; e < E_; ++e) if (z[e] > z[e1]) e1 = e;
  int e2 = (e1 == 0) ? 1 : 0;
#pragma unroll
  for (int e = 0; e < E_; ++e) if (e != e1 && z[e] > z[e2]) e2 = e;
  // renormalized top-2 softmax == sigmoid of the logit gap
  const float g1 = 1.f / (1.f + __expf(z[e2] - z[e1]));
  const float g2 = 1.f - g1;

  float d1 = 0.f, d2 = 0.f;
  const float* w1 = weh + e1 * H_ + lane * 16;
  const float* w2 = weh + e2 * H_ + lane * 16;
#pragma unroll
  for (int i = 0; i < 16; ++i) { d1 += c[i] * w1[i]; d2 += c[i] * w2[i]; }
  d1 += __shfl_xor(d1, 16); d1 += __shfl_xor(d1, 8); d1 += __shfl_xor(d1, 4);
  d1 += __shfl_xor(d1, 2);  d1 += __shfl_xor(d1, 1);
  d2 += __shfl_xor(d2, 16); d2 += __shfl_xor(d2, 8); d2 += __shfl_xor(d2, 4);
  d2 += __shfl_xor(d2, 2);  d2 += __shfl_xor(d2, 1);

  if (lane == 0) {
    const float contrib =
        (g1 * (d1 + beh[e1]) + g2 * (d2 + beh[e2])) * (1.0f / (float)S_);
    atomicAdd(out + (tok >> 10), contrib);  // tok / S_
  }
}

// ---------------------------------------------------------------------------
extern "C" void kernel_launch(void* const* d_in, const int* in_sizes, int n_in,
                              void* d_out, int out_size, void* d_ws, size_t ws_size,
                              hipStream_t stream) {
  (void)in_sizes; (void)n_in; (void)out_size; (void)ws_size;
  const float* x   = (const float*)d_in[0];
  const float* Wq  = (const float*)d_in[1];
  const float* bq  = (const float*)d_in[2];
  const float* Wk  = (const float*)d_in[3];
  const float* bk  = (const float*)d_in[4];
  const float* Wv  = (const float*)d_in[5];
  const float* bv  = (const float*)d_in[6];
  const float* Wiq = (const float*)d_in[7];
  const float* biq = (const float*)d_in[8];
  const float* Wik = (const float*)d_in[9];
  const float* bik = (const float*)d_in[10];
  const float* Wiv = (const float*)d_in[11];
  const float* biv = (const float*)d_in[12];
  const float* Wo  = (const float*)d_in[13];
  const float* bo  = (const float*)d_in[14];
  const float* Wg  = (const float*)d_in[15];
  const float* We  = (const float*)d_in[16];
  const float* be  = (const float*)d_in[17];
  const float* Wh  = (const float*)d_in[18];
  const float* bh  = (const float*)d_in[19];
  float* out = (float*)d_out;

  // Workspace: 4 x [B,S,H] fp32 (32MB each) + small folded-weight tail.
  char* ws = (char*)d_ws;
  const size_t BIG = (size_t)B_ * S_ * H_ * sizeof(float);
  float* bufQ = (float*)(ws + 0 * BIG);
  float* bufK = (float*)(ws + 1 * BIG);
  float* bufV = (float*)(ws + 2 * BIG);
  float* bufO = (float*)(ws + 3 * BIG);
  float* sm   = (float*)(ws + 4 * BIG);
  float* Wqc = sm; sm += F_ * H_;
  float* Wkc = sm; sm += F_ * H_;
  float* Wvc = sm; sm += F_ * H_;
  float* bqc = sm; sm += H_;
  float* bkc = sm; sm += H_;
  float* bvc = sm; sm += H_;
  float* weh = sm; sm += E_ * H_;
  float* beh = sm; sm += E_;
  float* o2  = bufK;  // K dead after attention; reuse for o@Wo+bo

  // 1) fold weights (tiny)
  const int tc = F_ * H_ + H_;
  combine_wb<<<dim3((tc + 255) / 256), dim3(256), 0, stream>>>(Wq, Wiq, bq, biq, Wqc, bqc);
  combine_wb<<<dim3((tc + 255) / 256), dim3(256), 0, stream>>>(Wk, Wik, bk, bik, Wkc, bkc);
  combine_wb<<<dim3((tc + 255) / 256), dim3(256), 0, stream>>>(Wv, Wiv, bv, biv, Wvc, bvc);
  const int tf = E_ * H_ + E_;
  fold_expert<<<dim3((tf + 255) / 256), dim3(256), 0, stream>>>(We, be, Wh, weh, beh);

  // 2) fused QKV projections: [B*S, F] @ [F, H] + bias  (WMMA fp32, K=64)
  const int tiles = (B_ * S_ / 16) * (H_ / 16);  // 32768 tiles, 8 per block
  dim3 gg(tiles / 8);
  gemm16<F_, H_><<<gg, 256, 0, stream>>>(x, Wqc, bqc, bufQ, B_ * S_);
  gemm16<F_, H_><<<gg, 256, 0, stream>>>(x, Wkc, bkc, bufK, B_ * S_);
  gemm16<F_, H_><<<gg, 256, 0, stream>>>(x, Wvc, bvc, bufV, B_ * S_);

  // 3) flash attention: B*NH*(S/16) = 4096 waves, 4 per block
  attn_flash<<<dim3((B_ * NH_ * (S_ / 16)) / 4), 128, 0, stream>>>(bufQ, bufK, bufV, bufO);

  // 4) o2 = o_att @ Wo + bo   (WMMA fp32, K=512)
  gemm16<H_, H_><<<gg, 256, 0, stream>>>(bufO, Wo, bo, o2, B_ * S_);

  // 5) gate + pooled head reduction
  init_out<<<dim3(1), dim3(32), 0, stream>>>(out, bh);
  moe_gate_reduce<<<dim3(B_ * S_ / 8), 256, 0, stream>>>(o2, Wg, weh, beh, out);
}